// DistanceLoss_1657857376401
// MI455X (gfx1250) — compile-verified
//
#include <hip/hip_runtime.h>
#include <hip/hip_bf16.h>

#define BB 8
#define CC 4
#define HH 384
#define WW 384
#define HW (HH*WW)            // 147456
#define N2 (BB*CC*HW)         // 4718592
#define NPIX (BB*HW)          // 1179648
#define BIGD 768.0f           // H+W sentinel, matches reference
#define CHPB (HW/64)          // 2304 chunks per batch image (exact)

#define K4_BLOCKS 288
#define K4_WAVES  (K4_BLOCKS*8)          // 2304 waves
#define K4_ITERS  ((NPIX/64)/K4_WAVES)   // 18432/2304 = 8 exact

static_assert(NPIX % 64 == 0, "chunking must be exact");
static_assert(HW % 64 == 0, "chunks never straddle a batch image");
static_assert((NPIX/64) % K4_WAVES == 0, "uniform trip count");

typedef float v2f __attribute__((ext_vector_type(2)));
typedef float v8f __attribute__((ext_vector_type(8)));

#if __has_builtin(__builtin_amdgcn_global_load_async_to_lds_b128) && \
    __has_builtin(__builtin_amdgcn_s_wait_asynccnt)
#define USE_ASYNC_LDS 1
// 16-byte int vectors in the address spaces the async builtin expects:
typedef int i32x4 __attribute__((vector_size(16)));
typedef __attribute__((address_space(1))) i32x4 gvec16;   // global (__device__)
typedef __attribute__((address_space(3))) i32x4 lvec16;   // LDS
#else
#define USE_ASYNC_LDS 0
#endif

// ---------------- Kernel 1: vertical 1D EDT (fwd+bwd scans), writes g^2 ----
__global__ void k1_vert(const int* __restrict__ targets, float* __restrict__ g2) {
    int col = blockIdx.x * blockDim.x + threadIdx.x;   // over B*C*W = 12288
    if (col >= BB * CC * WW) return;
    int b = col / (CC * WW);
    int rem = col - b * (CC * WW);
    int c = rem / WW;
    int w = rem - c * WW;
    const int* tcol = targets + (size_t)b * HW + w;                 // stride WW over h
    float* gcol = g2 + ((size_t)(b * CC + c) * HH) * WW + w;        // stride WW over h
    float d = BIGD;
    for (int h = 0; h < HH; ++h) {
        bool z = (tcol[(size_t)h * WW] == c);
        d = z ? 0.0f : fminf(d + 1.0f, BIGD);
        gcol[(size_t)h * WW] = d;                  // store forward distance
    }
    d = BIGD;
    for (int h = HH - 1; h >= 0; --h) {
        bool z = (tcol[(size_t)h * WW] == c);
        d = z ? 0.0f : fminf(d + 1.0f, BIGD);
        float f = fminf(d, gcol[(size_t)h * WW]);  // min(fwd,bwd)
        gcol[(size_t)h * WW] = f * f;              // store g^2
    }
}

// ---------------- Kernel 2: per-row exact envelope: dt2(y)=min_j g2(j)+(y-j)^2
// One workgroup per row; g2 row staged into LDS via CDNA5 async copy
// (GLOBAL_LOAD_ASYNC_TO_LDS_B128, ASYNCcnt); VALU-bound fmaf+min hot loop
// reading the staged row as float4 (ds_load_b128).
__global__ void __launch_bounds__(128) k2_horiz(float* __restrict__ g2) {
    __shared__ __align__(16) float row[WW];
    float* grow = g2 + (size_t)blockIdx.x * WW;    // blockIdx over B*C*H = 12288 rows

#if USE_ASYNC_LDS
    // 384 floats = 96 x 16B chunks; lanes 0..95 each issue one async B128 copy.
    if (threadIdx.x < 96) {
        __builtin_amdgcn_global_load_async_to_lds_b128(
            (gvec16*)(grow + threadIdx.x * 4),
            (lvec16*)(row + threadIdx.x * 4),
            /*offset=*/0, /*cpol=*/0);
    }
    __builtin_amdgcn_s_wait_asynccnt(0);   // per-wave: own async copies landed
    __syncthreads();                       // cross-wave visibility of LDS
#else
    for (int j = threadIdx.x; j < WW; j += 128) row[j] = grow[j];
    __syncthreads();
#endif

    float y0 = (float)threadIdx.x;
    float y1 = y0 + 128.0f;
    float y2 = y0 + 256.0f;
    float m0 = 3.0e38f, m1 = 3.0e38f, m2 = 3.0e38f;
    const float4* row4 = (const float4*)row;
    #pragma unroll 2
    for (int jj = 0; jj < WW / 4; ++jj) {
        float4 g = row4[jj];                       // one ds_load_b128 per 4 candidates
        float jf = (float)(4 * jj);
        float d;
        d = y0 - jf;          m0 = fminf(m0, fmaf(d, d, g.x));
        d = y0 - (jf + 1.f);  m0 = fminf(m0, fmaf(d, d, g.y));
        d = y0 - (jf + 2.f);  m0 = fminf(m0, fmaf(d, d, g.z));
        d = y0 - (jf + 3.f);  m0 = fminf(m0, fmaf(d, d, g.w));
        d = y1 - jf;          m1 = fminf(m1, fmaf(d, d, g.x));
        d = y1 - (jf + 1.f);  m1 = fminf(m1, fmaf(d, d, g.y));
        d = y1 - (jf + 2.f);  m1 = fminf(m1, fmaf(d, d, g.z));
        d = y1 - (jf + 3.f);  m1 = fminf(m1, fmaf(d, d, g.w));
        d = y2 - jf;          m2 = fminf(m2, fmaf(d, d, g.x));
        d = y2 - (jf + 1.f);  m2 = fminf(m2, fmaf(d, d, g.y));
        d = y2 - (jf + 2.f);  m2 = fminf(m2, fmaf(d, d, g.z));
        d = y2 - (jf + 3.f);  m2 = fminf(m2, fmaf(d, d, g.w));
    }
    // all global reads of this row finished before the barrier -> in-place is safe
    grow[threadIdx.x        ] = m0;
    grow[threadIdx.x + 128  ] = m1;
    grow[threadIdx.x + 256  ] = m2;
}

// ---------------- Kernel 3: per-(b,c) slice max of dt2 -> dmax = sqrt(max) ---
__global__ void k3_dmax(const float* __restrict__ dt2, float* __restrict__ dmax) {
    __shared__ float red[256];
    const float* p = dt2 + (size_t)blockIdx.x * HW;   // 32 slices
    float m = 0.0f;
    for (int i = threadIdx.x; i < HW; i += 256) m = fmaxf(m, p[i]);
    red[threadIdx.x] = m;
    __syncthreads();
    for (int off = 128; off > 0; off >>= 1) {
        if (threadIdx.x < off) red[threadIdx.x] = fmaxf(red[threadIdx.x], red[threadIdx.x + off]);
        __syncthreads();
    }
    if (threadIdx.x == 0) dmax[blockIdx.x] = sqrtf(red[0]);
}

// ---------------- per-pixel loss value: sum_c softmax_c * dist_c ------------
// ob/db are wave-uniform scalar bases; i = lane offset -> SADDR loads with
// scale_offset. ndm* are scalar (-dmax) values -> pure v_cndmask selects.
__device__ __forceinline__ float pixel_val(const float* __restrict__ ob,
                                           const int* __restrict__ tp,
                                           const float* __restrict__ db,
                                           float ndm0, float ndm1,
                                           float ndm2, float ndm3, int i) {
    float o0 = ob[i], o1 = ob[i + HW], o2 = ob[i + 2 * HW], o3 = ob[i + 3 * HW];
    float mx = fmaxf(fmaxf(o0, o1), fmaxf(o2, o3));
    float e0 = __expf(o0 - mx), e1 = __expf(o1 - mx);
    float e2 = __expf(o2 - mx), e3 = __expf(o3 - mx);
    float inv = __builtin_amdgcn_rcpf(e0 + e1 + e2 + e3);   // v_rcp_f32
    int t = tp[i];
    float s0 = __builtin_amdgcn_sqrtf(db[i]);               // v_sqrt_f32
    float s1 = __builtin_amdgcn_sqrtf(db[i + HW]);
    float s2 = __builtin_amdgcn_sqrtf(db[i + 2 * HW]);
    float s3 = __builtin_amdgcn_sqrtf(db[i + 3 * HW]);
    float d0 = (t == 0) ? ndm0 : s0;
    float d1 = (t == 1) ? ndm1 : s1;
    float d2 = (t == 2) ? ndm2 : s2;
    float d3 = (t == 3) ? ndm3 : s3;
    return (e0 * d0 + e1 * d1 + e2 * d2 + e3 * d3) * inv;
}

// ---------------- Kernel 4: loss accumulation via FP32 WMMA -----------------
// D(16x16,f32) += A(16x4,f32) x ones(4x16): folds 64 f32 values/instr, exact RNE.
// Wave-uniform chunk id forced scalar (readfirstlane) -> SALU addressing,
// s_load of dmax; fixed trip count keeps EXEC all-ones for WMMA.
__global__ void __launch_bounds__(256) k4_loss(const float* __restrict__ outputs,
                                               const int* __restrict__ targets,
                                               const float* __restrict__ dt2,
                                               const float* __restrict__ dmax,
                                               float* __restrict__ partials) {
    __shared__ float swt[8];
    int lane  = threadIdx.x & 31;
    int wid   = threadIdx.x >> 5;
    int gwave = blockIdx.x * 8 + wid;

    v8f acc = {0.f, 0.f, 0.f, 0.f, 0.f, 0.f, 0.f, 0.f};
    v2f bones; bones.x = 1.0f; bones.y = 1.0f;

    #pragma unroll 4
    for (int it = 0; it < K4_ITERS; ++it) {
        int ch  = __builtin_amdgcn_readfirstlane(gwave + it * K4_WAVES); // wave-uniform
        int b   = ch / CHPB;                       // scalar: batch image
        int rem = ch - b * CHPB;                   // scalar: chunk within image
        size_t base = (size_t)b * CC * HW + (size_t)rem * 64;
        const float* ob = outputs + base;
        const float* db = dt2 + base;
        const int*   tp = targets + (size_t)ch * 64;
        float ndm0 = -dmax[b * 4 + 0];             // uniform -> s_load
        float ndm1 = -dmax[b * 4 + 1];
        float ndm2 = -dmax[b * 4 + 2];
        float ndm3 = -dmax[b * 4 + 3];
        v2f a;
        a.x = pixel_val(ob, tp, db, ndm0, ndm1, ndm2, ndm3, lane);
        a.y = pixel_val(ob, tp, db, ndm0, ndm1, ndm2, ndm3, lane + 32);
        acc = __builtin_amdgcn_wmma_f32_16x16x4_f32(
            /*neg_a=*/false, a, /*neg_b=*/false, bones,
            /*c_mod=*/(short)0, acc, /*reuse_a=*/false, /*reuse_b=*/false);
    }

    // C/D layout: lane l<16 holds rows M=0..7 of column N=l; lanes>=16 hold M=8..15.
    // All columns identical (B==ones) -> wave total = halfsum + other halfsum.
    float s = acc[0] + acc[1] + acc[2] + acc[3] + acc[4] + acc[5] + acc[6] + acc[7];
    float wt = s + __shfl_xor(s, 16, 32);
    if (lane == 0) swt[wid] = wt;
    __syncthreads();
    if (threadIdx.x == 0) {
        float t = 0.0f;
        #pragma unroll
        for (int i = 0; i < 8; ++i) t += swt[i];
        partials[blockIdx.x] = t;
    }
}

// ---------------- Kernel 5: deterministic final reduction + scale ----------
__global__ void k5_final(const float* __restrict__ partials, int n, float* __restrict__ out) {
    __shared__ float red[256];
    float s = 0.0f;
    for (int i = threadIdx.x; i < n; i += 256) s += partials[i];
    red[threadIdx.x] = s;
    __syncthreads();
    for (int off = 128; off > 0; off >>= 1) {
        if (threadIdx.x < off) red[threadIdx.x] += red[threadIdx.x + off];
        __syncthreads();
    }
    // mean over B*C*H*W of (softmax*dist / C)  ->  total / (C * N2)
    if (threadIdx.x == 0) out[0] = red[0] * (1.0f / ((float)CC * (float)N2));
}

extern "C" void kernel_launch(void* const* d_in, const int* in_sizes, int n_in,
                              void* d_out, int out_size, void* d_ws, size_t ws_size,
                              hipStream_t stream) {
    const float* outputs = (const float*)d_in[0];   // [B,C,H,W] f32
    const int*   targets = (const int*)d_in[1];     // [B,H,W] i32
    float* out = (float*)d_out;                     // scalar f32

    float* g2       = (float*)d_ws;                 // N2 floats (g^2, then dt^2 in place)
    float* dmax     = g2 + N2;                      // 32 floats
    float* partials = dmax + 32;                    // K4_BLOCKS floats

    k1_vert <<< (BB * CC * WW + 255) / 256, 256, 0, stream >>> (targets, g2);
    k2_horiz<<< BB * CC * HH, 128, 0, stream >>> (g2);
    k3_dmax <<< BB * CC, 256, 0, stream >>> (g2, dmax);
    k4_loss <<< K4_BLOCKS, 256, 0, stream >>> (outputs, targets, g2, dmax, partials);
    k5_final<<< 1, 256, 0, stream >>> (partials, K4_BLOCKS, out);
}